// DIB_69861938037647
// MI455X (gfx1250) — compile-verified
//
#include <hip/hip_runtime.h>

// Attention over similar users:
//   scores[b,s,u] = sum_d cu[b,d] * sim[b,s,u,d]      (WMMA f32 16x16x4)
//   attn = softmax_u(mask ? -1e9 : scores)
//   out[b,s,d] = sum_u attn[u]*sim[b,s,u,d] + ci[b,s,d]
//
// Bandwidth-bound (~655MB of sim read once from HBM => ~28us floor @23.3TB/s).
// One wave32 per (b,s). Phase-1 reads are regular-temporal (so the phase-2
// re-read of the same 6.4KB tile hits WGP$/L2); phase-2 reads and the
// streaming ci/out traffic are non-temporal (last-use, never reused).

typedef float v2f __attribute__((ext_vector_type(2)));
typedef float v8f __attribute__((ext_vector_type(8)));

#define NEG_INF_F (-1e9f)

constexpr int Bn = 512;
constexpr int Sn = 200;
constexpr int Un = 50;
constexpr int Dn = 32;

__global__ __launch_bounds__(256) void DIB_attn_kernel(
    const float* __restrict__ cu,    // [B, D]
    const float* __restrict__ sim,   // [B, S, U, D]
    const float* __restrict__ ci,    // [B, S, D]
    const int*   __restrict__ mask,  // [B, S, U]  (1 = masked out)
    float* __restrict__ out)         // [B, S, D]
{
  const int lane = threadIdx.x & 31;
  const int ws   = blockIdx.x * 8 + (threadIdx.x >> 5);  // flat (b,s) index
  const int b    = ws / Sn;

  const float* simT = sim + (size_t)ws * (Un * Dn);  // this wave's 50x32 tile
  const float* cuB  = cu + b * Dn;
  const int*   mB   = mask + (size_t)ws * Un;

  const int hi  = lane >> 4;   // which half-wave (0/1)
  const int m15 = lane & 15;   // row within A / column within D

  // ---- Preload B operand: current-user vector broadcast into all 16 cols.
  // 32-bit 4x16 B layout (mirror of A): VGPR0 = {K=k0 | K=k0+2}, VGPR1 = {K=k0+1 | K=k0+3}.
  v2f bk[8];
#pragma unroll
  for (int k = 0; k < 8; ++k)
    bk[k] = *(const v2f*)(cuB + 4 * k + 2 * hi);

  // ---- Phase 1: scores via V_WMMA_F32_16X16X4_F32.
  // A layout per ISA: lanes 0-15 rows M=0..15 with {K=k0,k0+1}; lanes 16-31 with {K=k0+2,k0+3}.
  // That is exactly a b64 load of two consecutive d values per lane.
  v8f acc0 = {}, acc1 = {}, acc2 = {}, acc3 = {};
#pragma unroll
  for (int c = 0; c < 4; ++c) {
    int u = c * 16 + m15;
    u = (u > Un - 1) ? (Un - 1) : u;  // clamp phantom rows (results discarded)
    const float* arow = simT + (size_t)u * Dn + 2 * hi;
    v8f a_acc = (c == 0) ? acc0 : (c == 1) ? acc1 : (c == 2) ? acc2 : acc3;
#pragma unroll
    for (int k = 0; k < 8; ++k) {
      v2f a = *(const v2f*)(arow + 4 * k);
      a_acc = __builtin_amdgcn_wmma_f32_16x16x4_f32(
          /*neg_a=*/false, a, /*neg_b=*/false, bk[k],
          /*c_mod=*/(short)0, a_acc, /*reuse_a=*/false, /*reuse_b=*/false);
    }
    if (c == 0) acc0 = a_acc; else if (c == 1) acc1 = a_acc;
    else if (c == 2) acc2 = a_acc; else acc3 = a_acc;
  }

  // D layout: VGPR r holds scores[u = 16c + r + 8*hi], replicated across the 16
  // lanes of each half-wave. Each half-wave owns half the u-range.
  float sc[4][8];
  float mloc = NEG_INF_F;
#pragma unroll
  for (int c = 0; c < 4; ++c) {
    const v8f a_acc = (c == 0) ? acc0 : (c == 1) ? acc1 : (c == 2) ? acc2 : acc3;
#pragma unroll
    for (int r = 0; r < 8; ++r) {
      const int u = 16 * c + r + 8 * hi;        // runtime (depends on hi)
      const int uc = (u < Un) ? u : (Un - 1);   // always-in-bounds address
      const int mv = mB[uc];                    // unconditional load: EXEC stays uniform
      const bool dead = (mv != 0) | (u >= Un);  // masked user or phantom row
      sc[c][r] = dead ? NEG_INF_F : a_acc[r];
      mloc = fmaxf(mloc, sc[c][r]);
    }
  }
  // merge max across half-waves
  const float mAll = fmaxf(mloc, __shfl_xor(mloc, 16, 32));

  float ex[4][8];
  float sloc = 0.0f;
#pragma unroll
  for (int c = 0; c < 4; ++c) {
#pragma unroll
    for (int r = 0; r < 8; ++r) {
      const int u = 16 * c + r + 8 * hi;
      // phantom u must contribute 0 (so the all-masked case gives uniform 1/50)
      float e = (u < Un) ? __expf(sc[c][r] - mAll) : 0.0f;
      ex[c][r] = e;
      sloc += e;
    }
  }
  const float sAll = sloc + __shfl_xor(sloc, 16, 32);
  const float inv = 1.0f / sAll;

  float at[4][8], atw[4][8];
#pragma unroll
  for (int c = 0; c < 4; ++c) {
#pragma unroll
    for (int r = 0; r < 8; ++r) {
      at[c][r]  = ex[c][r] * inv;                    // own half's attn[u]
      atw[c][r] = __shfl_xor(at[c][r], 16, 32);      // other half's attn[u^8]
    }
  }

  // ---- Phase 2: out[d] = sum_u attn[u] * tile[u][d]; lane = d. Last use of
  // the tile: non-temporal so lines retire from cache immediately after.
  float o = 0.0f;
#pragma unroll
  for (int u = 0; u < Un; ++u) {
    const int c = u >> 4;
    const int r = u & 7;
    const int h = (u >> 3) & 1;  // which half-wave natively holds attn[u]
    float bu;
    if (h == 0) bu = (hi == 0) ? at[c][r] : atw[c][r];
    else        bu = (hi == 0) ? atw[c][r] : at[c][r];
    float t = __builtin_nontemporal_load(simT + (size_t)u * Dn + lane);
    o = fmaf(bu, t, o);
  }

  const size_t oidx = (size_t)ws * Dn + lane;
  const float cival = __builtin_nontemporal_load(ci + oidx);
  __builtin_nontemporal_store(o + cival, out + oidx);
}

extern "C" void kernel_launch(void* const* d_in, const int* in_sizes, int n_in,
                              void* d_out, int out_size, void* d_ws, size_t ws_size,
                              hipStream_t stream) {
  const float* cu  = (const float*)d_in[0];  // current_user_embedding [B,D]
  const float* sim = (const float*)d_in[1];  // similar_user_embedding [B,S,U,D]
  const float* ci  = (const float*)d_in[2];  // current_item_embedding [B,S,D]
  const int*   msk = (const int*)d_in[3];    // mask [B,S,U]
  float* out = (float*)d_out;                // [B,S,D]

  const int nWaves = Bn * Sn;                // 102400 (b,s) pairs
  dim3 grid(nWaves / 8);                     // 8 waves (256 threads) per block
  dim3 block(256);
  DIB_attn_kernel<<<grid, block, 0, stream>>>(cu, sim, ci, msk, out);
}